// GlobalModel_6279242186981
// MI455X (gfx1250) — compile-verified
//
#include <hip/hip_runtime.h>
#include <hip/hip_bf16.h>

typedef __attribute__((ext_vector_type(2))) float v2f;
typedef __attribute__((ext_vector_type(8))) float v8f;

#define N_NODES 50000
#define N_EDGES 1600000
#define NB      512
#define D_NODE  256
#define D_EDGE  128
#define D_U     128
#define D_IN    512      // D_U + D_NODE + D_EDGE
#define HID     512
#define D_OUT   128
#define EPS     1e-5f

// workspace offsets (in floats)
#define OFF_XSUM  0                       // 512*256
#define OFF_ESUM  131072                  // 512*128
#define OFF_XCNT  196608                  // 512
#define OFF_ECNT  197120                  // 512
#define OFF_CSUM  197632                  // 512
#define OFF_CSQ   198144                  // 512
#define ZERO_N    198656
#define OFF_SCALE 198656                  // 512
#define OFF_SHIFT 199168                  // 512
#define OFF_H     199680                  // 512*512 assembled input
#define OFF_H1    461824                  // 512*512 pre-BN hidden

__device__ __forceinline__ void gatomic_add(float* p, float v) {
  __hip_atomic_fetch_add(p, v, __ATOMIC_RELAXED, __HIP_MEMORY_SCOPE_AGENT);
}
__device__ __forceinline__ void latomic_add(float* p, float v) {
  __hip_atomic_fetch_add(p, v, __ATOMIC_RELAXED, __HIP_MEMORY_SCOPE_WORKGROUP);
}

__global__ void k_zero(float* __restrict__ ws, int n) {
  int i = blockIdx.x * blockDim.x + threadIdx.x;
  if (i < n) ws[i] = 0.f;
}

// ---- node scatter-mean: batch is SORTED -> per-block register segmented sum.
// thread t owns feature t (coalesced row reads); one global atomic per segment.
__global__ void k_node_aggr(const float* __restrict__ x,
                            const long long* __restrict__ batch,
                            float* __restrict__ ws) {
  const int t   = threadIdx.x;                       // feature 0..255
  const int per = (N_NODES + gridDim.x - 1) / gridDim.x;
  const int n0  = blockIdx.x * per;
  const int n1  = min(n0 + per, N_NODES);
  if (n0 >= n1) return;
  float acc = 0.f;
  int   g   = (int)batch[n0];
  int   run = 0;
  for (int n = n0; n < n1; ++n) {
    const int bg = (int)batch[n];
    if (bg != g) {
      gatomic_add(&ws[OFF_XSUM + g * D_NODE + t], acc);
      if (t == 0) gatomic_add(&ws[OFF_XCNT + g], (float)run);
      acc = 0.f; run = 0; g = bg;
    }
    acc += x[(size_t)n * D_NODE + t];
    ++run;
  }
  gatomic_add(&ws[OFF_XSUM + g * D_NODE + t], acc);
  if (t == 0) gatomic_add(&ws[OFF_XCNT + g], (float)run);
}

// ---- edge scatter-mean: full 512x128 f32 accumulator table in LDS (256KB+2KB,
// only possible with CDNA5's 320KB WGP LDS). One wave handles one edge row:
// lane loads float4 (global_load_b128), ds_add_f32 into table, flush once.
__global__ void k_edge_aggr(const float* __restrict__ ea,
                            const long long* __restrict__ col,
                            const long long* __restrict__ batch,
                            float* __restrict__ ws) {
  extern __shared__ float lds[];                     // [NB*D_EDGE] sums + [NB] counts
  const int tid = threadIdx.x;
  for (int i = tid; i < NB * D_EDGE + NB; i += blockDim.x) lds[i] = 0.f;
  __syncthreads();

  const int lane = tid & 31;                         // wave32
  const int wv   = tid >> 5;
  const int nw   = blockDim.x >> 5;
  const int per  = (N_EDGES + gridDim.x - 1) / gridDim.x;
  const int e0   = blockIdx.x * per;
  const int e1   = min(e0 + per, N_EDGES);

  for (int e = e0 + wv; e < e1; e += nw) {
    const int c = (int)col[e];
    const int g = (int)batch[c];
    const float4 v = *(const float4*)(ea + (size_t)e * D_EDGE + lane * 4);
    float* dst = &lds[g * D_EDGE + lane * 4];
    latomic_add(dst + 0, v.x);
    latomic_add(dst + 1, v.y);
    latomic_add(dst + 2, v.z);
    latomic_add(dst + 3, v.w);
    if (lane == 0) latomic_add(&lds[NB * D_EDGE + g], 1.f);
  }
  __syncthreads();

  for (int i = tid; i < NB * D_EDGE + NB; i += blockDim.x) {
    const float v = lds[i];
    if (v != 0.f) {
      if (i < NB * D_EDGE) gatomic_add(&ws[OFF_ESUM + i], v);
      else                 gatomic_add(&ws[OFF_ECNT + (i - NB * D_EDGE)], v);
    }
  }
}

// ---- h = concat(u, x_sum/max(cnt,1), e_sum/max(cnt,1))
__global__ void k_assemble(const float* __restrict__ u, float* __restrict__ ws) {
  const int b  = blockIdx.x;
  const float xc = fmaxf(ws[OFF_XCNT + b], 1.f);
  const float ec = fmaxf(ws[OFF_ECNT + b], 1.f);
  for (int j = threadIdx.x; j < D_IN; j += blockDim.x) {
    float v;
    if (j < D_U)               v = u[b * D_U + j];
    else if (j < D_U + D_NODE) v = ws[OFF_XSUM + b * D_NODE + (j - D_U)] / xc;
    else                       v = ws[OFF_ESUM + b * D_EDGE + (j - D_U - D_NODE)] / ec;
    ws[OFF_H + b * D_IN + j] = v;
  }
}

// ---- GEMM1: h[512,512] @ W1[512,512] + b1 via V_WMMA_F32_16X16X4_F32.
// One wave per 16x16 tile; fused column sum/sumsq for BN stats.
// Layouts (ISA 7.12.2): A/B VGPR j holds K = k + j + 2*(lane>>4);
// C/D VGPR r holds row r + 8*(lane>>4), col = lane&15.
__global__ void k_gemm1(const float* __restrict__ W1,
                        const float* __restrict__ b1,
                        float* __restrict__ ws) {
  const int lane = threadIdx.x & 31;
  const int wv   = threadIdx.x >> 5;
  const int tile = blockIdx.x * 8 + wv;
  const int tM   = tile >> 5;                        // 32 tiles in N
  const int tN   = tile & 31;
  const int hi   = lane >> 4;
  const int lo   = lane & 15;
  const int m    = tM * 16 + lo;
  const int n    = tN * 16 + lo;
  const float* A = ws + OFF_H;

  v8f acc = {};
  for (int k = 0; k < D_IN; k += 4) {
    const int ka = k + 2 * hi;
    const v2f a = *(const v2f*)(A + m * D_IN + ka);  // K=ka, ka+1 (8B aligned)
    v2f b;
    b[0] = W1[ka * HID + n];
    b[1] = W1[(ka + 1) * HID + n];
    acc = __builtin_amdgcn_wmma_f32_16x16x4_f32(
        false, a, false, b, (short)0, acc, false, false);
  }

  const float bias = b1[n];
  float s = 0.f, s2 = 0.f;
  for (int r = 0; r < 8; ++r) {
    const float v   = acc[r] + bias;
    const int   row = tM * 16 + r + 8 * hi;
    ws[OFF_H1 + row * HID + n] = v;
    s += v; s2 += v * v;
  }
  gatomic_add(&ws[OFF_CSUM + n], s);
  gatomic_add(&ws[OFF_CSQ  + n], s2);
}

// ---- BN fold: scale = gamma*rsqrt(var+eps), shift = beta - mean*scale
__global__ void k_bnprep(const float* __restrict__ gamma,
                         const float* __restrict__ beta,
                         float* __restrict__ ws) {
  const int k = blockIdx.x * blockDim.x + threadIdx.x;
  if (k >= HID) return;
  const float mean = ws[OFF_CSUM + k] * (1.f / NB);
  const float var  = ws[OFF_CSQ  + k] * (1.f / NB) - mean * mean;
  const float inv  = rsqrtf(var + EPS);
  const float sc   = gamma[k] * inv;
  ws[OFF_SCALE + k] = sc;
  ws[OFF_SHIFT + k] = beta[k] - mean * sc;
}

// ---- GEMM2: relu(h1*scale+shift)[512,512] @ W2[512,128] + b2 via WMMA f32.
__global__ void k_gemm2(const float* __restrict__ W2,
                        const float* __restrict__ b2,
                        const float* __restrict__ ws,
                        float* __restrict__ out) {
  const float* H1    = ws + OFF_H1;
  const float* scale = ws + OFF_SCALE;
  const float* shift = ws + OFF_SHIFT;
  const int lane = threadIdx.x & 31;
  const int wv   = threadIdx.x >> 5;
  const int tile = blockIdx.x * 8 + wv;
  const int tM   = tile >> 3;                        // 8 tiles in N
  const int tN   = tile & 7;
  const int hi   = lane >> 4;
  const int lo   = lane & 15;
  const int m    = tM * 16 + lo;
  const int n    = tN * 16 + lo;

  v8f acc = {};
  for (int k = 0; k < HID; k += 4) {
    const int ka = k + 2 * hi;
    v2f a, b;
    a[0] = fmaxf(H1[m * HID + ka]     * scale[ka]     + shift[ka],     0.f);
    a[1] = fmaxf(H1[m * HID + ka + 1] * scale[ka + 1] + shift[ka + 1], 0.f);
    b[0] = W2[ka * D_OUT + n];
    b[1] = W2[(ka + 1) * D_OUT + n];
    acc = __builtin_amdgcn_wmma_f32_16x16x4_f32(
        false, a, false, b, (short)0, acc, false, false);
  }

  const float bias = b2[n];
  for (int r = 0; r < 8; ++r) {
    const int row = tM * 16 + r + 8 * hi;
    out[row * D_OUT + n] = acc[r] + bias;
  }
}

extern "C" void kernel_launch(void* const* d_in, const int* in_sizes, int n_in,
                              void* d_out, int out_size, void* d_ws, size_t ws_size,
                              hipStream_t stream) {
  const float*     x     = (const float*)d_in[0];
  const long long* eidx  = (const long long*)d_in[1];   // int64 [2,E]
  const float*     ea    = (const float*)d_in[2];
  const float*     u     = (const float*)d_in[3];
  const long long* batch = (const long long*)d_in[4];   // int64 [N], sorted
  const float*     W1    = (const float*)d_in[5];
  const float*     b1    = (const float*)d_in[6];
  const float*     gamma = (const float*)d_in[7];
  const float*     beta  = (const float*)d_in[8];
  const float*     W2    = (const float*)d_in[9];
  const float*     b2    = (const float*)d_in[10];
  float* out = (float*)d_out;
  float* ws  = (float*)d_ws;
  const long long* col = eidx + N_EDGES;                // edge_index[1]

  k_zero<<<(ZERO_N + 255) / 256, 256, 0, stream>>>(ws, ZERO_N);
  k_node_aggr<<<256, 256, 0, stream>>>(x, batch, ws);
  k_edge_aggr<<<256, 256, (NB * D_EDGE + NB) * sizeof(float), stream>>>(ea, col, batch, ws);
  k_assemble<<<NB, 256, 0, stream>>>(u, ws);
  k_gemm1<<<128, 256, 0, stream>>>(W1, b1, ws);         // 1024 tiles, 8 waves/block
  k_bnprep<<<2, 256, 0, stream>>>(gamma, beta, ws);
  k_gemm2<<<32, 256, 0, stream>>>(W2, b2, ws, out);     // 256 tiles
}